// MyCRF_47390669144633
// MI455X (gfx1250) — compile-verified
//
#include <hip/hip_runtime.h>
#include <math.h>

#define BB 64
#define SS 512
#define FF 1024
#define NL 50
#define NSTART 48
#define NSTOP 49
#define NPAD 64

typedef __attribute__((ext_vector_type(2))) float v2f;
typedef __attribute__((ext_vector_type(8))) float v8f;

// ---------------------------------------------------------------------------
// Pad W (F x 50) -> Wp (F x 64), zero-filled columns 50..63
// ---------------------------------------------------------------------------
__global__ void crf_padw_kernel(const float* __restrict__ W, float* __restrict__ Wp) {
    int i = blockIdx.x * blockDim.x + threadIdx.x;   // over F*NPAD
    int k = i >> 6;
    int n = i & 63;
    Wp[i] = (n < NL) ? W[k * NL + n] : 0.0f;
}

// ---------------------------------------------------------------------------
// logits[row, n] = features[row, :] @ Wp[:, n] + b[n]
// One wave computes a 32-row x 64-col tile: two 16-row M-tiles share every
// B fragment (4 N-tiles x f32 WMMA), so each K-chunk costs
// 2 A-loads (B64) + 8 B-loads (B32) for 8 v_wmma_f32_16x16x4_f32.
// A 16x4 f32 fragment layout: M = lane%16, lanes 0-15 hold K={k,k+1},
// lanes 16-31 hold K={k+2,k+3}. B 4x16 mirrors it with N = lane%16.
// ---------------------------------------------------------------------------
__global__ void crf_gemm_kernel(const float* __restrict__ feat,
                                const float* __restrict__ Wp,
                                const float* __restrict__ bias,
                                float* __restrict__ logits) {
    const int wave = threadIdx.x >> 5;
    const int lane = threadIdx.x & 31;
    const int row0 = (blockIdx.x * 4 + wave) * 32;   // 32 rows per wave
    const int m = lane & 15;
    const int kh = lane >> 4;          // 0 or 1: which K pair this half-wave owns

    v8f accA0 = {}, accA1 = {}, accA2 = {}, accA3 = {};   // rows row0..row0+15
    v8f accB0 = {}, accB1 = {}, accB2 = {}, accB3 = {};   // rows row0+16..row0+31
    const float* arow0 = feat + (size_t)(row0 + m) * FF + 2 * kh;
    const float* arow1 = arow0 + (size_t)16 * FF;

    #pragma unroll 2
    for (int k = 0; k < FF; k += 4) {
        v2f a0 = *(const v2f*)(arow0 + k);
        v2f a1 = *(const v2f*)(arow1 + k);
        const float* bp = Wp + (size_t)(k + 2 * kh) * NPAD + m;
        v2f bf;
        bf.x = bp[0];       bf.y = bp[NPAD];
        accA0 = __builtin_amdgcn_wmma_f32_16x16x4_f32(false, a0, false, bf, (short)0, accA0, false, false);
        accB0 = __builtin_amdgcn_wmma_f32_16x16x4_f32(false, a1, false, bf, (short)0, accB0, false, false);
        bf.x = bp[16];      bf.y = bp[NPAD + 16];
        accA1 = __builtin_amdgcn_wmma_f32_16x16x4_f32(false, a0, false, bf, (short)0, accA1, false, false);
        accB1 = __builtin_amdgcn_wmma_f32_16x16x4_f32(false, a1, false, bf, (short)0, accB1, false, false);
        bf.x = bp[32];      bf.y = bp[NPAD + 32];
        accA2 = __builtin_amdgcn_wmma_f32_16x16x4_f32(false, a0, false, bf, (short)0, accA2, false, false);
        accB2 = __builtin_amdgcn_wmma_f32_16x16x4_f32(false, a1, false, bf, (short)0, accB2, false, false);
        bf.x = bp[48];      bf.y = bp[NPAD + 48];
        accA3 = __builtin_amdgcn_wmma_f32_16x16x4_f32(false, a0, false, bf, (short)0, accA3, false, false);
        accB3 = __builtin_amdgcn_wmma_f32_16x16x4_f32(false, a1, false, bf, (short)0, accB3, false, false);
    }

    // C/D layout: VGPR v holds row M=v (lanes 0-15) or M=8+v (lanes 16-31); N = lane%16.
    v8f accs[2][4] = {{accA0, accA1, accA2, accA3}, {accB0, accB1, accB2, accB3}};
    #pragma unroll
    for (int t = 0; t < 4; ++t) {
        int col = t * 16 + m;
        float bv = (col < NL) ? bias[col] : 0.0f;
        #pragma unroll
        for (int h = 0; h < 2; ++h) {
            #pragma unroll
            for (int v = 0; v < 8; ++v) {
                int row = row0 + 16 * h + v + 8 * kh;
                logits[(size_t)row * NPAD + col] = accs[h][t][v] + bv;
            }
        }
    }
}

// ---------------------------------------------------------------------------
// Per-batch forward scan (logsumexp recurrence) + gold path score.
// One block of 64 threads per batch element. transitions cached in LDS with
// row stride 51 (gcd(51,64)=1 -> conflict-free); per-thread transition row
// hoisted into registers; alpha double-buffered, one barrier per step.
// alpha[cur][j] reads are same-address across lanes -> LDS broadcast.
// ---------------------------------------------------------------------------
__global__ void crf_scan_kernel(const float* __restrict__ logits,
                                const int* __restrict__ labels,
                                const int* __restrict__ lens,
                                const float* __restrict__ trans,
                                float* __restrict__ diff) {
    __shared__ float t_lds[NL * 51];
    __shared__ float alpha[2][NPAD];
    __shared__ float red[64];

    const int b = blockIdx.x;
    const int tid = threadIdx.x;
    const int len = lens[b];
    const int* lab = labels + (size_t)b * SS;
    const float* lg = logits + (size_t)b * SS * NPAD;

    for (int i = tid; i < NL * NL; i += 64)
        t_lds[(i / NL) * 51 + (i % NL)] = trans[i];
    alpha[0][tid] = (tid == NSTART) ? 0.0f : -10000.0f;
    alpha[1][tid] = -10000.0f;
    __syncthreads();

    // ---- gold score (parallel over time) ----
    float gold = 0.0f;
    for (int t = tid; t < len; t += 64)
        gold += lg[(size_t)t * NPAD + lab[t]];
    for (int t = tid; t <= len; t += 64) {
        int prev = (t == 0)   ? NSTART : lab[t - 1];
        int next = (t == len) ? NSTOP  : lab[t];
        gold += t_lds[next * 51 + prev];
    }
    red[tid] = gold;
    __syncthreads();
    float gold_total = 0.0f;
    if (tid == 0)
        for (int i = 0; i < 64; ++i) gold_total += red[i];
    __syncthreads();   // red is reused below

    // ---- forward recurrence ----
    float trow[NL];
    if (tid < NL) {
        #pragma unroll
        for (int j = 0; j < NL; ++j) trow[j] = t_lds[tid * 51 + j];
    }

    int cur = 0;
    for (int t = 0; t < len; ++t) {
        float anew = 0.0f;
        if (tid < NL) {
            float mx = -3.0e38f;
            #pragma unroll
            for (int j = 0; j < NL; ++j)
                mx = fmaxf(mx, trow[j] + alpha[cur][j]);
            float s = 0.0f;
            #pragma unroll
            for (int j = 0; j < NL; ++j)
                s += __expf(trow[j] + alpha[cur][j] - mx);
            anew = lg[(size_t)t * NPAD + tid] + mx + __logf(s);
        }
        if (tid < NL) alpha[cur ^ 1][tid] = anew;
        __syncthreads();
        cur ^= 1;
    }

    // ---- final logsumexp: alpha + transitions[STOP, :] ----
    float fin = -3.0e38f;
    if (tid < NL) fin = alpha[cur][tid] + t_lds[NSTOP * 51 + tid];
    red[tid] = fin;
    __syncthreads();
    if (tid == 0) {
        float mx = -3.0e38f;
        for (int i = 0; i < NL; ++i) mx = fmaxf(mx, red[i]);
        float s = 0.0f;
        for (int i = 0; i < NL; ++i) s += __expf(red[i] - mx);
        float norm = mx + __logf(s);
        diff[b] = norm - gold_total;
    }
}

// ---------------------------------------------------------------------------
// mean over 64 per-batch diffs
// ---------------------------------------------------------------------------
__global__ void crf_mean_kernel(const float* __restrict__ diff, float* __restrict__ out) {
    __shared__ float sm[BB];
    sm[threadIdx.x] = diff[threadIdx.x];
    __syncthreads();
    if (threadIdx.x == 0) {
        float s = 0.0f;
        for (int i = 0; i < BB; ++i) s += sm[i];
        out[0] = s / (float)BB;
    }
}

extern "C" void kernel_launch(void* const* d_in, const int* in_sizes, int n_in,
                              void* d_out, int out_size, void* d_ws, size_t ws_size,
                              hipStream_t stream) {
    const float* features = (const float*)d_in[0];   // (B,S,F) f32
    const int*   labels   = (const int*)d_in[1];     // (B,S)  i32
    const int*   lens     = (const int*)d_in[2];     // (B,)   i32
    const float* W        = (const float*)d_in[3];   // (F,50) f32
    const float* bias     = (const float*)d_in[4];   // (50,)  f32
    const float* trans    = (const float*)d_in[5];   // (50,50) f32

    float* logits = (float*)d_ws;                          // 32768*64 f32 = 8 MB
    float* Wp     = logits + (size_t)BB * SS * NPAD;       // 1024*64 f32
    float* diff   = Wp + (size_t)FF * NPAD;                // 64 f32

    crf_padw_kernel<<<(FF * NPAD) / 256, 256, 0, stream>>>(W, Wp);
    crf_gemm_kernel<<<(BB * SS) / (32 * 4), 128, 0, stream>>>(features, Wp, bias, logits);
    crf_scan_kernel<<<BB, 64, 0, stream>>>(logits, labels, lens, trans, diff);
    crf_mean_kernel<<<1, BB, 0, stream>>>(diff, (float*)d_out);
}